// Attention_44178033607060
// MI455X (gfx1250) — compile-verified
//
#include <hip/hip_runtime.h>
#include <cmath>
#include <stdint.h>

typedef __attribute__((ext_vector_type(2))) float v2f;
typedef __attribute__((ext_vector_type(8))) float v8f;

#define B_  32
#define T_  4096
#define H_  256
#define D_  512
#define W4H (4 * H_)   // row stride of W_attn = 1024

// ---------------------------------------------------------------------------
// Kernel 1: hp[b][h] = hidden[b,:] . W_attn[h, 0:512]  + b_attn[h]
// ---------------------------------------------------------------------------
__global__ __launch_bounds__(256)
void hproj_kernel(const float* __restrict__ hidden,
                  const float* __restrict__ W,
                  const float* __restrict__ battn,
                  float* __restrict__ hp) {
  __shared__ float sh[D_];
  const int b = blockIdx.x, tid = threadIdx.x;
  for (int i = tid; i < D_; i += 256) sh[i] = hidden[b * D_ + i];
  __syncthreads();
  const float* wr = W + (size_t)tid * W4H;   // W_h row h = tid
  float s = 0.f;
  #pragma unroll 4
  for (int d = 0; d < D_; d += 4) {
    float4 w4 = *(const float4*)(wr + d);
    s += sh[d] * w4.x + sh[d + 1] * w4.y + sh[d + 2] * w4.z + sh[d + 3] * w4.w;
  }
  hp[b * H_ + tid] = s + battn[tid];
}

// ---------------------------------------------------------------------------
// Kernel 2 (fused, WMMA f32 16x16x4, async double-buffered LDS staging):
//   e_proj tile = enc[b, t0:t0+64, :] @ W_e.T   (K = 512, chunked by 16)
//   scores[b,t] = sum_h tanh(e_proj + hp[b][h]) * v[h]
// Block: 256 thr = 8 waves. Wave w: t-tile (w&3), N-half (w>>2) = 8 h-tiles.
// Staging uses GLOBAL_LOAD_ASYNC_TO_LDS_B128 (ASYNCcnt) with 2 LDS buffers:
// chunk c+1 streams in while WMMAs consume chunk c. Rows padded to 20 floats
// (80 B, 16B-aligned) => conflict-free ds b64 fragment reads.
// ---------------------------------------------------------------------------
__global__ __launch_bounds__(256, 1)
void scores_kernel(const float* __restrict__ enc,
                   const float* __restrict__ W,
                   const float* __restrict__ hp,
                   const float* __restrict__ v,
                   float* __restrict__ scores) {
  __shared__ float sW[2][256][20];   // [buf][n][k]  (40 KB)
  __shared__ float sA[2][64][20];    // [buf][r][k]  (10 KB)
  __shared__ float sScore[64];

  const int tid  = threadIdx.x;
  const int lane = tid & 31;
  const int lo   = lane & 15;     // M (A) / N (B) index
  const int hi   = lane >> 4;     // selects K pair {0,1} vs {2,3}
  const int wv   = tid >> 5;
  const int tt   = wv & 3;        // t-tile within block
  const int nh   = wv >> 2;       // N-half (0/1)

  const int b   = blockIdx.y;
  const int t0b = blockIdx.x * 64;

  if (tid < 64) sScore[tid] = 0.f;

  const float* Wbase = W;                          // SGPR base for W async loads
  const float* Ebase = enc + (size_t)b * T_ * D_;  // SGPR base for enc async loads

  // W_e transfers: 1024 b128 per chunk, 4 per thread. id = g*256+tid,
  // n = id>>2 (row), kg = id&3 (16B group within the 16-float chunk).
  unsigned wvo[4];
  unsigned wlds[2][4];
  #pragma unroll
  for (int g = 0; g < 4; ++g) {
    const int id = g * 256 + tid;
    const int n  = id >> 2, kg = id & 3;
    wvo[g]     = (unsigned)(n * W4H + D_ + kg * 4) * 4u;       // bytes, k0 = 0
    wlds[0][g] = (unsigned)(uintptr_t)&sW[0][n][kg * 4];
    wlds[1][g] = (unsigned)(uintptr_t)&sW[1][n][kg * 4];
  }
  // enc transfers: 256 b128 per chunk, 1 per thread. r = tid>>2, kq = tid&3.
  const int r = tid >> 2, kq = tid & 3;
  const unsigned avo = (unsigned)(((t0b + r) * D_) + kq * 4) * 4u;  // bytes
  const unsigned alds[2] = { (unsigned)(uintptr_t)&sA[0][r][kq * 4],
                             (unsigned)(uintptr_t)&sA[1][r][kq * 4] };

  v8f acc[8] = {};   // 8 h-tiles of 16x16 f32 accumulators

  // ---- prologue: stage chunk 0 into buffer 0 ----
  {
    #pragma unroll
    for (int g = 0; g < 4; ++g)
      asm volatile("global_load_async_to_lds_b128 %0, %1, %2"
                   :: "v"(wlds[0][g]), "v"(wvo[g]), "s"(Wbase) : "memory");
    asm volatile("global_load_async_to_lds_b128 %0, %1, %2"
                 :: "v"(alds[0]), "v"(avo), "s"(Ebase) : "memory");
  }
  asm volatile("s_wait_asynccnt 0x0" ::: "memory");
  __syncthreads();

  for (int c = 0; c < 32; ++c) {           // 32 chunks x 16 K = 512
    const int buf = c & 1;
    if (c + 1 < 32) {                      // uniform: EXEC stays all-ones
      const unsigned kb = (unsigned)(c + 1) * 16u * 4u;   // byte advance in K
      #pragma unroll
      for (int g = 0; g < 4; ++g)
        asm volatile("global_load_async_to_lds_b128 %0, %1, %2"
                     :: "v"(wlds[buf ^ 1][g]), "v"(wvo[g] + kb), "s"(Wbase) : "memory");
      asm volatile("global_load_async_to_lds_b128 %0, %1, %2"
                   :: "v"(alds[buf ^ 1]), "v"(avo + kb), "s"(Ebase) : "memory");
    }

    #pragma unroll
    for (int ks = 0; ks < 4; ++ks) {
      const int k = ks * 4 + hi * 2;
      // A fragment (16x4 f32): lane lo holds row lo, K = {k, k+1}
      float2 af = *(const float2*)&sA[buf][tt * 16 + lo][k];
      v2f a; a.x = af.x; a.y = af.y;
      #pragma unroll
      for (int nt = 0; nt < 8; ++nt) {
        // B fragment (4x16): B[k][n] = W_e[n][k]
        float2 bf = *(const float2*)&sW[buf][(nh * 8 + nt) * 16 + lo][k];
        v2f bb; bb.x = bf.x; bb.y = bf.y;
        acc[nt] = __builtin_amdgcn_wmma_f32_16x16x4_f32(
            false, a, false, bb, (short)0, acc[nt], false, false);
      }
    }

    asm volatile("s_wait_asynccnt 0x0" ::: "memory");
    __syncthreads();
  }

  // Epilogue: tanh + dot with v, reduced over this wave's 128 h values.
  // C layout: VGPR j = rows {j (lanes0-15), j+8 (lanes16-31)}, col = lo.
  float sc[8];
  #pragma unroll
  for (int j = 0; j < 8; ++j) sc[j] = 0.f;
  #pragma unroll
  for (int nt = 0; nt < 8; ++nt) {
    const int ng = (nh * 8 + nt) * 16 + lo;
    const float hb = hp[b * H_ + ng];
    const float vv = v[ng];
    #pragma unroll
    for (int j = 0; j < 8; ++j)
      sc[j] += tanhf(acc[nt][j] + hb) * vv;
  }
  #pragma unroll
  for (int j = 0; j < 8; ++j)
    atomicAdd(&sScore[tt * 16 + hi * 8 + j], sc[j]);   // ds_add_f32
  __syncthreads();
  if (tid < 64) scores[(size_t)b * T_ + t0b + tid] = sScore[tid];
}

// ---------------------------------------------------------------------------
// Kernel 3: in-place softmax over T per batch
// ---------------------------------------------------------------------------
__global__ __launch_bounds__(256)
void softmax_kernel(float* __restrict__ attn) {
  __shared__ float red[256];
  const int b = blockIdx.x, tid = threadIdx.x;
  float* s = attn + (size_t)b * T_;
  float vals[16];
  float m = -3.4e38f;
  #pragma unroll
  for (int i = 0; i < 16; ++i) { vals[i] = s[tid + i * 256]; m = fmaxf(m, vals[i]); }
  red[tid] = m; __syncthreads();
  for (int off = 128; off > 0; off >>= 1) {
    if (tid < off) red[tid] = fmaxf(red[tid], red[tid + off]);
    __syncthreads();
  }
  m = red[0]; __syncthreads();
  float sum = 0.f;
  #pragma unroll
  for (int i = 0; i < 16; ++i) { vals[i] = expf(vals[i] - m); sum += vals[i]; }
  red[tid] = sum; __syncthreads();
  for (int off = 128; off > 0; off >>= 1) {
    if (tid < off) red[tid] += red[tid + off];
    __syncthreads();
  }
  const float inv = 1.f / red[0];
  #pragma unroll
  for (int i = 0; i < 16; ++i) s[tid + i * 256] = vals[i] * inv;
}

// ---------------------------------------------------------------------------
// Kernel 4: context[b][d] = sum_t attn[b][t] * enc[b][t][d]
// ---------------------------------------------------------------------------
__global__ __launch_bounds__(256)
void context_kernel(const float* __restrict__ enc,
                    const float* __restrict__ attn,
                    float* __restrict__ ctx) {
  __shared__ float swt[T_];        // 16 KB
  __shared__ float sred[3][64];
  const int b   = blockIdx.y;
  const int d0  = blockIdx.x * 64;
  const int tid = threadIdx.x;
  const int dl  = tid & 63;
  const int tq  = tid >> 6;        // 0..3 : t phase
  for (int i = tid; i < T_; i += 256) swt[i] = attn[(size_t)b * T_ + i];
  __syncthreads();
  float acc = 0.f;
  const float* base = enc + (size_t)b * T_ * D_ + d0 + dl;
  for (int t = tq; t < T_; t += 4)
    acc += swt[t] * base[(size_t)t * D_];
  if (tq > 0) sred[tq - 1][dl] = acc;
  __syncthreads();
  if (tq == 0) {
    acc += sred[0][dl] + sred[1][dl] + sred[2][dl];
    ctx[(size_t)b * D_ + d0 + dl] = acc;
  }
}

// ---------------------------------------------------------------------------
extern "C" void kernel_launch(void* const* d_in, const int* in_sizes, int n_in,
                              void* d_out, int out_size, void* d_ws, size_t ws_size,
                              hipStream_t stream) {
  const float* hidden = (const float*)d_in[0];   // (32, 512)
  const float* enc    = (const float*)d_in[1];   // (32, 4096, 512)
  const float* W      = (const float*)d_in[2];   // (256, 1024)
  const float* battn  = (const float*)d_in[3];   // (256,)
  const float* v      = (const float*)d_in[4];   // (256,)

  float* out  = (float*)d_out;
  float* ctx  = out;                // context: B*D = 16384 floats
  float* attn = out + B_ * D_;      // attn_weights: B*T = 131072 floats
  float* hp   = (float*)d_ws;       // h_proj + bias: B*H floats (32 KB)

  hipLaunchKernelGGL(hproj_kernel,   dim3(B_),          dim3(256), 0, stream,
                     hidden, W, battn, hp);
  hipLaunchKernelGGL(scores_kernel,  dim3(T_ / 64, B_), dim3(256), 0, stream,
                     enc, W, hp, v, attn);
  hipLaunchKernelGGL(softmax_kernel, dim3(B_),          dim3(256), 0, stream,
                     attn);
  hipLaunchKernelGGL(context_kernel, dim3(D_ / 64, B_), dim3(256), 0, stream,
                     enc, attn, ctx);
}